// MultiScaleDTW_23983097381099
// MI455X (gfx1250) — compile-verified
//
#include <hip/hip_runtime.h>
#include <hip/hip_bf16.h>

typedef __attribute__((ext_vector_type(2))) float v2f;
typedef __attribute__((ext_vector_type(8))) float v8f;

#define BIGV 1e30f
#define L 1024
#define D 128
#define LD1 1025   // dtw row length

// ---------------------------------------------------------------------------
// Kernel 0: squared-norm of every row of s1 and s2.  n1/n2: [16*1024] each.
// ---------------------------------------------------------------------------
__global__ __launch_bounds__(256) void dtw_norms_kernel(
    const float* __restrict__ s1, const float* __restrict__ s2,
    float* __restrict__ n1, float* __restrict__ n2) {
  int idx = blockIdx.x * blockDim.x + threadIdx.x;   // 0 .. 16*1024-1
  if (idx >= 16 * L) return;
  const float4* p1 = (const float4*)(s1 + (long)idx * D);
  const float4* p2 = (const float4*)(s2 + (long)idx * D);
  float a = 0.f, b = 0.f;
#pragma unroll 8
  for (int i = 0; i < D / 4; ++i) {
    float4 v = p1[i];
    a += v.x * v.x + v.y * v.y + v.z * v.z + v.w * v.w;
    float4 w = p2[i];
    b += w.x * w.x + w.y * w.y + w.z * w.z + w.w * w.w;
  }
  n1[idx] = a;
  n2[idx] = b;
}

// ---------------------------------------------------------------------------
// Kernel 1: cost matrix via V_WMMA_F32_16X16X4_F32.
// One wave computes one 16x16 tile of cost[b] = sqrt(max(n1+n2-2*s1.s2^T,eps)),
// and stores it into the dtw output at [b][i+1][j+1] (overwritten in-place by
// the DTW kernel afterwards).
//
// Fragment layouts (ISA 7.12.2, wave32):
//   A 16x4 f32 : lane l -> row M=l%16; VGPR0/1 hold K = 2*(l/16) + {0,1}
//   B 4x16 f32 : lane l -> col N=l%16; VGPR0/1 hold K = 2*(l/16) + {0,1}
//   C/D 16x16  : VGPR r -> row M = r + 8*(l/16), col N = l%16
// ---------------------------------------------------------------------------
__global__ __launch_bounds__(256) void dtw_cost_wmma_kernel(
    const float* __restrict__ s1, const float* __restrict__ s2,
    const float* __restrict__ n1, const float* __restrict__ n2,
    float* __restrict__ dtw) {
  const int wave = threadIdx.x >> 5;
  const int lane = threadIdx.x & 31;
  const long gwave = (long)blockIdx.x * (blockDim.x >> 5) + wave;  // 0..65535
  const int b   = (int)(gwave >> 12);        // 4096 tiles per batch
  const int tid = (int)(gwave & 4095);
  const int ti  = tid >> 6;                  // 64 row tiles
  const int tj  = tid & 63;                  // 64 col tiles
  const int half = lane >> 4;                // 0 or 1
  const int lm   = lane & 15;

  const float2* Ap = (const float2*)(s1 + ((long)b * L + ti * 16 + lm) * D);
  const float2* Bp = (const float2*)(s2 + ((long)b * L + tj * 16 + lm) * D);

  v8f acc = {};
#pragma unroll
  for (int k0 = 0; k0 < D; k0 += 4) {
    const int kh = (k0 >> 1) + half;         // float2 index: K = k0 + 2*half
    float2 av = Ap[kh];
    float2 bv = Bp[kh];
    v2f a; a.x = av.x; a.y = av.y;
    v2f bb; bb.x = bv.x; bb.y = bv.y;
    acc = __builtin_amdgcn_wmma_f32_16x16x4_f32(
        /*neg_a=*/false, a, /*neg_b=*/false, bb,
        /*c_mod=*/(short)0, acc, /*reuse_a=*/false, /*reuse_b=*/false);
  }

  const int col = tj * 16 + lm;
  const float nn2 = n2[b * L + col];
  float* base = dtw + (long)b * LD1 * LD1;
#pragma unroll
  for (int r = 0; r < 8; ++r) {
    const int row = ti * 16 + r + 8 * half;
    float d2 = n1[b * L + row] + nn2 - 2.0f * acc[r];
    d2 = fmaxf(d2, 1e-12f);
    base[(long)(row + 1) * LD1 + (col + 1)] = sqrtf(d2);
  }
}

// ---------------------------------------------------------------------------
// Kernel 2: DTW. One workgroup (1024 threads = 32 waves) per batch element.
// Row recurrence r[j] = c[j] + min(m[j], r[j-1]) solved as an inclusive
// min-plus associative scan (Hillis-Steele in LDS, 10 steps).
// Reads each cost row from the dtw slot and overwrites it with the dtw row.
// ---------------------------------------------------------------------------
__global__ __launch_bounds__(1024) void dtw_scan_kernel(
    float* __restrict__ dtw, const float* __restrict__ scale_w,
    float* __restrict__ out_sim) {
  __shared__ float prev[LD1];
  __shared__ float sa[L];
  __shared__ float sb[L];

  const int t = threadIdx.x;                 // 0..1023, handles dtw column t+1
  const int b = blockIdx.x;
  float* base = dtw + (long)b * LD1 * LD1;

  // dtw row 0 = [0, BIG, BIG, ...]
  base[t] = (t == 0) ? 0.0f : BIGV;
  prev[t] = (t == 0) ? 0.0f : BIGV;
  if (t == 1023) { base[L] = BIGV; prev[L] = BIGV; }
  __syncthreads();

  float yb = BIGV;
  for (int i = 1; i <= L; ++i) {
    float* rowp = base + (long)i * LD1;
    const float c = rowp[t + 1];                       // cost[i-1][t]
    const float m = fminf(prev[t], prev[t + 1]);       // min(diag, up)
    float ya = c;
    yb = c + m;
    sa[t] = ya;
    sb[t] = yb;
    __syncthreads();
#pragma unroll
    for (int d = 1; d < L; d <<= 1) {
      float xa = 0.f, xb = 0.f;
      const bool act = (t >= d);
      if (act) { xa = sa[t - d]; xb = sb[t - d]; }
      __syncthreads();
      if (act) {
        yb = fminf(yb, ya + xb);   // min(b_y, a_y + b_x)
        ya = ya + xa;              // a_x + a_y
        sa[t] = ya;
        sb[t] = yb;
      }
      __syncthreads();
    }
    rowp[t + 1] = yb;              // dtw[i][t+1]
    prev[t + 1] = yb;
    if (t == 0) { rowp[0] = BIGV; prev[0] = BIGV; }
    __syncthreads();
  }

  if (t == 1023) {
    // dtw[b][L][L] / (L1+L2); softmax over a single weight is identically 1.
    const float dist = yb / (float)(2 * L);
    const float sim = 1.0f / (1.0f + dist);
    (void)scale_w;
    out_sim[b] = sim;
  }
}

// ---------------------------------------------------------------------------
extern "C" void kernel_launch(void* const* d_in, const int* in_sizes, int n_in,
                              void* d_out, int out_size, void* d_ws, size_t ws_size,
                              hipStream_t stream) {
  const float* s1 = (const float*)d_in[0];           // [16,1024,128]
  const float* s2 = (const float*)d_in[1];           // [16,1024,128]
  const float* sw = (const float*)d_in[2];           // [1]

  float* out  = (float*)d_out;
  float* sim  = out;                                 // [16]
  float* dtw  = out + 16;                            // [16,1025,1025]

  float* n1 = (float*)d_ws;                          // [16*1024]
  float* n2 = n1 + 16 * L;                           // [16*1024]

  // 0) row norms
  dtw_norms_kernel<<<(16 * L + 255) / 256, 256, 0, stream>>>(s1, s2, n1, n2);

  // 1) cost matrix via fp32 WMMA: 16*64*64 = 65536 tiles, 8 waves/block
  dtw_cost_wmma_kernel<<<65536 / 8, 256, 0, stream>>>(s1, s2, n1, n2, dtw);

  // 2) DTW: one 32-wave workgroup per batch element
  dtw_scan_kernel<<<16, 1024, 0, stream>>>(dtw, sw, sim);
}